// DiscreteCRFConv_72662256714587
// MI455X (gfx1250) — compile-verified
//
#include <hip/hip_runtime.h>
#include <hip/hip_bf16.h>
#include <stdint.h>

typedef float v2f __attribute__((ext_vector_type(2)));
typedef float v8f __attribute__((ext_vector_type(8)));

#define N_NODES 50000
#define DEG     16
#define NCH     16          // n_channels (classes)
#define ECH     64          // e_channels
#define HID     64          // hidden
#define NK      5           // kernels
#define KH      (NK*HID)    // 320
#define MTILES  (N_NODES/16)   // 3125 exactly
#define JTILES  (KH/16)        // 20
#define FSTRIDE 66             // padded LDS row stride (words) to kill bank conflicts

// ---------------------------------------------------------------- u = -log(p)
__global__ void neg_log_kernel(const float* __restrict__ p,
                               float* __restrict__ u, int n) {
    int i = blockIdx.x * blockDim.x + threadIdx.x;
    if (i < n) u[i] = -logf(p[i]);
}

// ------------------------------------------------- fp[n, k*64+h] = f @ Fk
// One block (640 thr = 20 waves) per 16-node row; the 16x64 f-tile is staged in
// LDS once and shared by all 20 j-tiles; each wave runs 16 chained f32 WMMAs.
__global__ void fp_gemm_kernel(const float* __restrict__ f,
                               const float* __restrict__ Fk,
                               float* __restrict__ fp) {
    __shared__ float ftile[16 * FSTRIDE];       // padded 16x64 A tile
    const int nb = blockIdx.x * 16;             // node base

    for (int i = threadIdx.x; i < 16 * ECH; i += blockDim.x) {
        int r = i >> 6, cc = i & 63;
        ftile[r * FSTRIDE + cc] = f[(size_t)(nb + r) * ECH + cc];
    }
    __syncthreads();

    const int lane = threadIdx.x & 31;
    const int tj   = threadIdx.x >> 5;          // wave id = j-tile 0..19
    const int jb   = tj * 16;
    const int k    = jb / HID;
    const int h0   = jb % HID;
    const int g    = lane >> 4;
    const int c    = lane & 15;

    const float* arow  = &ftile[c * FSTRIDE];   // A: M = lane%16, from LDS
    const float* bbase = Fk + (size_t)k * (ECH * HID) + h0 + c;

    v8f acc = {0.f, 0.f, 0.f, 0.f, 0.f, 0.f, 0.f, 0.f};
#pragma unroll
    for (int c0 = 0; c0 < ECH; c0 += 4) {
        v2f a = *(const v2f*)(arow + c0 + 2 * g);       // ds_load_b64
        v2f b;
        b.x = bbase[(size_t)(c0 + 2 * g + 0) * HID];
        b.y = bbase[(size_t)(c0 + 2 * g + 1) * HID];
        acc = __builtin_amdgcn_wmma_f32_16x16x4_f32(false, a, false, b,
                                                    (short)0, acc, false, false);
    }
    float* drow = fp + jb + c;
#pragma unroll
    for (int r = 0; r < 8; ++r)
        drow[(size_t)(nb + r + 8 * g) * KH] = acc[r];
}

// ---------------------------------- w[e] = sum_k Wk[k]*exp(-||fp[col]-fp[row]||^2)
// One block (512 thr = 16 waves) per node; fp[row] staged into LDS via the CDNA5
// async path (GLOBAL_LOAD_ASYNC_TO_LDS_B128 + s_wait_asynccnt); wave per edge.
__global__ void edge_w_kernel(const float* __restrict__ fp,
                              const int* __restrict__ col,
                              const float* __restrict__ Wk,
                              float* __restrict__ w) {
    __shared__ float frow[KH];
    const int node = blockIdx.x;

    // 320 floats = 80 x b128; threads 0..79 each issue one async copy.
    if (threadIdx.x < KH / 4) {
        // low 32 bits of a generic LDS pointer == LDS byte offset (ISA 10.2)
        uint32_t lds_off = (uint32_t)(uintptr_t)(&frow[threadIdx.x * 4]);
        uint64_t gaddr   = (uint64_t)(uintptr_t)(fp + (size_t)node * KH +
                                                 threadIdx.x * 4);
        asm volatile("global_load_async_to_lds_b128 %0, %1, off"
                     :: "v"(lds_off), "v"(gaddr) : "memory");
    }
    asm volatile("s_wait_asynccnt 0x0" ::: "memory");
    __syncthreads();

    const int wid  = threadIdx.x >> 5;          // edge slot 0..15
    const int lane = threadIdx.x & 31;
    const int e    = node * DEG + wid;
    const int cn   = col[e];
    const float* grow = fp + (size_t)cn * KH;

    float acc = 0.f;
#pragma unroll
    for (int k = 0; k < NK; ++k) {
        v2f gv = *(const v2f*)(grow + k * HID + lane * 2);
        float dx = gv.x - frow[k * HID + lane * 2 + 0];
        float dy = gv.y - frow[k * HID + lane * 2 + 1];
        float s  = dx * dx + dy * dy;           // lane partial over 2 of 64 h's
#pragma unroll
        for (int off = 16; off >= 1; off >>= 1)
            s += __shfl_xor(s, off, 32);
        acc += expf(-s) * Wk[k];
    }
    if (lane == 0) w[e] = acc;
}

// ------------------- one mean-field step: q' = softmax(-u - (segsum(q[col]*w)) @ C)
// One wave per 16-node tile (segments contiguous: node n owns edges 16n..16n+15).
// Messages -> per-wave LDS 16x16 tile -> 4 chained f32 WMMAs vs C -> softmax.
__global__ void crf_step_kernel(const float* __restrict__ qin,
                                const int* __restrict__ col,
                                const float* __restrict__ w,
                                const float* __restrict__ u,
                                const float* __restrict__ Cmat,
                                float* __restrict__ qout) {
    __shared__ float sT[8][NCH * NCH];          // 16x16 msg tile per wave
    const int lane = threadIdx.x & 31;
    const int wid  = threadIdx.x >> 5;
    const int tile = blockIdx.x * (blockDim.x >> 5) + wid;
    if (tile >= MTILES) return;                 // wave-uniform guard (EXEC all-1)
    const int nb = tile * 16;
    const int g  = lane >> 4;
    const int c  = lane & 15;
    float* s = sT[wid];

    // message accumulation: s[m][c] = sum_{j} qin[col[16(nb+m)+j], c] * w[...]
    for (int m = 0; m < 16; ++m) {
        const int ebase = (nb + m) * DEG;
        float acc = 0.f;
#pragma unroll
        for (int j = 0; j < DEG / 2; ++j) {     // each half-wave covers 8 edges
            int e  = ebase + 2 * j + g;
            int cl = col[e];
            acc += qin[(size_t)cl * NCH + c] * w[e];
        }
        acc += __shfl_xor(acc, 16, 32);         // combine the two halves
        if (g == 0) s[m * NCH + c] = acc;
    }
    // Same-wave LDS RAW: compiler inserts s_wait_dscnt before reuse.

    // D = S @ C via 4 chained V_WMMA_F32_16X16X4_F32 (K = 16 channels)
    v8f acc = {0.f, 0.f, 0.f, 0.f, 0.f, 0.f, 0.f, 0.f};
#pragma unroll
    for (int kk = 0; kk < NCH; kk += 4) {
        v2f a, b;
        a.x = s[c * NCH + kk + 2 * g + 0];      // A: M = lane%16, K = kk+2g..+1
        a.y = s[c * NCH + kk + 2 * g + 1];
        b.x = Cmat[(kk + 2 * g + 0) * NCH + c]; // B: K rows of C, N = lane%16
        b.y = Cmat[(kk + 2 * g + 1) * NCH + c];
        acc = __builtin_amdgcn_wmma_f32_16x16x4_f32(false, a, false, b,
                                                    (short)0, acc, false, false);
    }

    // softmax over channels: row m of D lives across 16 lanes in vgpr r
#pragma unroll
    for (int r = 0; r < 8; ++r) {
        int m   = nb + r + 8 * g;
        float t = -u[(size_t)m * NCH + c] - acc[r];
        float mx = t;
        mx = fmaxf(mx, __shfl_xor(mx, 1, 32));
        mx = fmaxf(mx, __shfl_xor(mx, 2, 32));
        mx = fmaxf(mx, __shfl_xor(mx, 4, 32));
        mx = fmaxf(mx, __shfl_xor(mx, 8, 32));
        float ex = expf(t - mx);
        float sm = ex;
        sm += __shfl_xor(sm, 1, 32);
        sm += __shfl_xor(sm, 2, 32);
        sm += __shfl_xor(sm, 4, 32);
        sm += __shfl_xor(sm, 8, 32);
        qout[(size_t)m * NCH + c] = ex / sm;
    }
}

// ---------------------------------------------------------------------- launch
extern "C" void kernel_launch(void* const* d_in, const int* in_sizes, int n_in,
                              void* d_out, int out_size, void* d_ws, size_t ws_size,
                              hipStream_t stream) {
    const float* p   = (const float*)d_in[0];
    const float* f   = (const float*)d_in[1];
    const int*   col = (const int*)d_in[2];
    /* d_in[3] = row: implicit (row[e] = e / DEG), unused */
    const float* Fk  = (const float*)d_in[4];
    const float* Wk  = (const float*)d_in[5];
    const float* Cm  = (const float*)d_in[6];
    float* out = (float*)d_out;
    (void)in_sizes; (void)n_in; (void)out_size; (void)ws_size;

    char* ws = (char*)d_ws;
    size_t off = 0;
    auto carve = [&](size_t bytes) -> float* {
        float* r = (float*)(ws + off);
        off = (off + bytes + 255) & ~(size_t)255;
        return r;
    };
    float* fp = carve(sizeof(float) * (size_t)N_NODES * KH);   // 64.0 MB
    float* w  = carve(sizeof(float) * (size_t)N_NODES * DEG);  //  3.2 MB
    float* u  = carve(sizeof(float) * (size_t)N_NODES * NCH);  //  3.2 MB
    float* q0 = carve(sizeof(float) * (size_t)N_NODES * NCH);  //  3.2 MB
    float* q1 = carve(sizeof(float) * (size_t)N_NODES * NCH);  //  3.2 MB

    // u = -log(p)
    {
        int n = N_NODES * NCH;
        neg_log_kernel<<<(n + 255) / 256, 256, 0, stream>>>(p, u, n);
    }
    // fp = f @ Fk  (WMMA f32; one block per node-tile, 20 waves share LDS A-tile)
    fp_gemm_kernel<<<MTILES, 640, 0, stream>>>(f, Fk, fp);

    // edge weights (async-to-LDS staging)
    edge_w_kernel<<<N_NODES, 512, 0, stream>>>(fp, col, Wk, w);

    // 5 mean-field steps, ping-pong buffers, last step writes d_out
    int tb = (MTILES + 7) / 8;
    crf_step_kernel<<<tb, 256, 0, stream>>>(p,  col, w, u, Cm, q0);
    crf_step_kernel<<<tb, 256, 0, stream>>>(q0, col, w, u, Cm, q1);
    crf_step_kernel<<<tb, 256, 0, stream>>>(q1, col, w, u, Cm, q0);
    crf_step_kernel<<<tb, 256, 0, stream>>>(q0, col, w, u, Cm, q1);
    crf_step_kernel<<<tb, 256, 0, stream>>>(q1, col, w, u, Cm, out);
}